// Highpass_68418829025362
// MI455X (gfx1250) — compile-verified
//
#include <hip/hip_runtime.h>
#include <cmath>

// ---------------- problem constants ----------------
#define TT    480000          // samples per row
#define BB    64              // batch rows
#define CHUNK 256             // samples per wave
#define NCH   (TT / CHUNK)    // 1875 chunks per row
#define PI_D  3.14159265358979323846

typedef float v2f __attribute__((ext_vector_type(2)));
typedef float v8f __attribute__((ext_vector_type(8)));

// All uniform filter-derived constants, computed on host in double.
struct Params {
  float a1, a2, b1, b2;
  float al14, al15, bt14, bt15;
  float M1[4], M2[4], M4[4], M8[4];           // sub-block map powers {a,b,c,d}
  float C1[4], C2[4], C4[4], C8[4], C16[4];   // chunk map (M^16) powers
  float al[16], bt[16];                       // homogeneous basis
};

// ---------------- init: build per-lane Toeplitz A-operand table ----------
// aop[lane*8 + i*2 + v] = T[m][4i + 2*half + v] = h_{m-j} (0 if j > m)
__global__ __launch_bounds__(32) void biquad_init(
    float *__restrict__ aop,
    double B0, double B1, double B2, double A1, double A2) {
  const int lane = threadIdx.x;
  const int n = lane & 15, half = lane >> 4;
  double hh[16];
  hh[0] = B0;
  hh[1] = B1 - A1 * hh[0];
  hh[2] = B2 - A1 * hh[1] - A2 * hh[0];
#pragma unroll
  for (int k = 3; k < 16; ++k) hh[k] = -A1 * hh[k - 1] - A2 * hh[k - 2];
#pragma unroll
  for (int i = 0; i < 4; ++i) {
#pragma unroll
    for (int v = 0; v < 2; ++v) {
      int j = 4 * i + 2 * half + v;
      int idx = n - j;
      float t = 0.f;
#pragma unroll
      for (int k = 0; k < 16; ++k) t = (idx == k) ? (float)hh[k] : t;
      aop[lane * 8 + i * 2 + v] = t;
    }
  }
}

// ---------------- passes 1 & 3: WMMA zero-state + intra-wave affine scan ----
__global__ __launch_bounds__(256) void biquad_pass13(
    const float *__restrict__ x, float *__restrict__ y,
    float2 *__restrict__ dchunk, const float2 *__restrict__ zin,
    const float *__restrict__ aopTab, int pass3, Params P) {
  const int lane = threadIdx.x & 31;
  const int wid = (blockIdx.x * blockDim.x + threadIdx.x) >> 5; // 1 wave/chunk
  const int r = wid / NCH;
  const int c = wid % NCH;
  const int n = lane & 15;     // sub-block index (column of B/D)
  const int half = lane >> 4;  // K / M half selector

  // ---- A operand: two b128 loads from the precomputed per-lane table ----
  const float4 *at = (const float4 *)aopTab + lane * 2;
  float4 t0 = at[0], t1 = at[1];
  v2f Aop0 = {t0.x, t0.y}, Aop1 = {t0.z, t0.w};
  v2f Aop2 = {t1.x, t1.y}, Aop3 = {t1.z, t1.w};

  // ---- B operand: column n = sub-block n, B[k][n] = x[n*16 + k] ----
  const float *xb = x + (size_t)r * TT + (size_t)c * CHUNK;
  const int off = n * 16 + half * 2;
  v2f Bop[4];
#pragma unroll
  for (int i = 0; i < 4; ++i) {
    float2 t = *(const float2 *)(xb + off + 4 * i);
    Bop[i][0] = t.x;
    Bop[i][1] = t.y;
  }

  // ---- D = T * X : 16x16x16 via four v_wmma_f32_16x16x4_f32 ----
  v8f acc = {0.f, 0.f, 0.f, 0.f, 0.f, 0.f, 0.f, 0.f};
  acc = __builtin_amdgcn_wmma_f32_16x16x4_f32(false, Aop0, false, Bop[0],
                                              (short)0, acc, false, false);
  acc = __builtin_amdgcn_wmma_f32_16x16x4_f32(false, Aop1, false, Bop[1],
                                              (short)0, acc, false, false);
  acc = __builtin_amdgcn_wmma_f32_16x16x4_f32(false, Aop2, false, Bop[2],
                                              (short)0, acc, false, false);
  acc = __builtin_amdgcn_wmma_f32_16x16x4_f32(false, Aop3, false, Bop[3],
                                              (short)0, acc, false, false);

  // zero-state tail (rows 14,15) of sub-block n lives in lane n+16, VGPR 6/7
  float yz15 = __shfl(acc[7], 16 + n);
  float yz14 = __shfl(acc[6], 16 + n);

  // boundary x's for this sub-block (x_{-1}, x_{-2}); zeros at row start
  float xm1, xm2;
  if (c == 0 && n == 0) {
    xm1 = 0.f;
    xm2 = 0.f;
  } else {
    float2 t = *(const float2 *)(xb + n * 16 - 2);
    xm2 = t.x;
    xm1 = t.y;
  }

  float sx = P.b1 * xm1 + P.b2 * xm2; // x-part of delta0
  float tx = P.b2 * xm1;              // raw x-part of delta1
  float u1x = tx - P.a1 * sx;         // x-part of delta1 (after -a1*delta0)

  // affine-scan payload: w_n = yzs tail + x-boundary contribution
  float wzx = yz15 + P.al15 * sx + P.bt15 * u1x; // y15 component
  float wzy = yz14 + P.al14 * sx + P.bt14 * u1x; // y14 component

  float2 zi = make_float2(0.f, 0.f);
  if (pass3) zi = zin[wid];
  if (n == 0) { // fold chunk-incoming state into sub-block 0
    wzx += P.M1[0] * zi.x + P.M1[1] * zi.y;
    wzy += P.M1[2] * zi.x + P.M1[3] * zi.y;
  }

  // Kogge-Stone affine scan across 16 sub-blocks: z_n = M z_{n-1} + w_n
  float zxv = wzx, zyv = wzy;
  auto ks = [&](float q00, float q01, float q10, float q11, int d) {
    float sxv = __shfl(zxv, lane - d);
    float syv = __shfl(zyv, lane - d);
    float nx = zxv + q00 * sxv + q01 * syv;
    float ny = zyv + q10 * sxv + q11 * syv;
    if (n >= d) { zxv = nx; zyv = ny; }
  };
  ks(P.M1[0], P.M1[1], P.M1[2], P.M1[3], 1);
  ks(P.M2[0], P.M2[1], P.M2[2], P.M2[3], 2);
  ks(P.M4[0], P.M4[1], P.M4[2], P.M4[3], 4);
  ks(P.M8[0], P.M8[1], P.M8[2], P.M8[3], 8);

  if (!pass3) {
    if (lane == 15) dchunk[wid] = make_float2(zxv, zyv); // zero-state chunk tail
    return;
  }

  // exclusive state for this sub-block -> correction (delta0, delta1)
  float pzx = __shfl(zxv, lane - 1);
  float pzy = __shfl(zyv, lane - 1);
  if (n == 0) { pzx = zi.x; pzy = zi.y; }

  float d0 = sx - P.a1 * pzx - P.a2 * pzy;
  float d1 = tx - P.a2 * pzx - P.a1 * d0;

  // y_k = yzs_k + al_k*d0 + bt_k*d1 ; lane half selects rows 0-7 vs 8-15
  float outv[8];
#pragma unroll
  for (int q = 0; q < 8; ++q) {
    float ak = half ? P.al[q + 8] : P.al[q];
    float bk = half ? P.bt[q + 8] : P.bt[q];
    outv[q] = acc[q] + ak * d0 + bk * d1;
  }
  float *yb = y + (size_t)r * TT + (size_t)c * CHUNK + n * 16 + half * 8;
  *(float4 *)(yb + 0) = make_float4(outv[0], outv[1], outv[2], outv[3]);
  *(float4 *)(yb + 4) = make_float4(outv[4], outv[5], outv[6], outv[7]);
}

// ---------------- pass 2: per-row affine scan over 1875 chunk maps ----------
__global__ __launch_bounds__(32) void biquad_pass2(
    const float2 *__restrict__ dchunk, float2 *__restrict__ zin, Params P) {
  const int lane = threadIdx.x;
  const int r = blockIdx.x;

  float cx = 0.f, cy = 0.f; // carry: state entering first chunk of this group
  const int ngrp = (NCH + 31) / 32;
  for (int g = 0; g < ngrp; ++g) {
    int idx = g * 32 + lane;
    float wx = 0.f, wy = 0.f;
    if (idx < NCH) {
      float2 t = dchunk[r * NCH + idx];
      wx = t.x;
      wy = t.y;
    }
    float ocx = cx, ocy = cy;
    if (lane == 0) { // fold carry into lane 0's payload
      wx += P.C1[0] * cx + P.C1[1] * cy;
      wy += P.C1[2] * cx + P.C1[3] * cy;
    }
    float zx = wx, zy = wy;
    auto ks = [&](float q00, float q01, float q10, float q11, int d) {
      float sxv = __shfl(zx, lane - d);
      float syv = __shfl(zy, lane - d);
      float nx = zx + q00 * sxv + q01 * syv;
      float ny = zy + q10 * sxv + q11 * syv;
      if (lane >= d) { zx = nx; zy = ny; }
    };
    ks(P.C1[0], P.C1[1], P.C1[2], P.C1[3], 1);
    ks(P.C2[0], P.C2[1], P.C2[2], P.C2[3], 2);
    ks(P.C4[0], P.C4[1], P.C4[2], P.C4[3], 4);
    ks(P.C8[0], P.C8[1], P.C8[2], P.C8[3], 8);
    ks(P.C16[0], P.C16[1], P.C16[2], P.C16[3], 16);
    float pzx = __shfl(zx, lane - 1);
    float pzy = __shfl(zy, lane - 1);
    if (lane == 0) { pzx = ocx; pzy = ocy; }
    if (idx < NCH) zin[r * NCH + idx] = make_float2(pzx, pzy);
    cx = __shfl(zx, 31);
    cy = __shfl(zy, 31);
  }
}

// ---------------- host ----------------
struct HM22 { double a, b, c, d; };
static HM22 hmul(const HM22 &x, const HM22 &y) {
  return {x.a * y.a + x.b * y.c, x.a * y.b + x.b * y.d,
          x.c * y.a + x.d * y.c, x.c * y.b + x.d * y.d};
}
static void put(float *dst, const HM22 &m) {
  dst[0] = (float)m.a; dst[1] = (float)m.b;
  dst[2] = (float)m.c; dst[3] = (float)m.d;
}

extern "C" void kernel_launch(void *const *d_in, const int *in_sizes, int n_in,
                              void *d_out, int out_size, void *d_ws,
                              size_t ws_size, hipStream_t stream) {
  (void)in_sizes; (void)n_in; (void)out_size; (void)ws_size;
  const float *x = (const float *)d_in[0];
  float *y = (float *)d_out;
  float2 *dchunk = (float2 *)d_ws;                 // 64*1875 float2
  float2 *zin = dchunk + (size_t)BB * NCH;         // 64*1875 float2
  float *aopTab = (float *)(zin + (size_t)BB * NCH); // 32*8 floats (1KB)

  // RBJ highpass coefficients in double, normalized by a0
  const double Qf = 0.7071067811865476;
  const double w0 = 2.0 * PI_D * 100.0 / 16000.0;
  const double alp = sin(w0) / (2.0 * Qf);
  const double cw = cos(w0);
  double b0 = (1.0 + cw) * 0.5, b1 = -(1.0 + cw), b2 = b0;
  double a0 = 1.0 + alp, a1 = -2.0 * cw, a2 = 1.0 - alp;
  b0 /= a0; b1 /= a0; b2 /= a0; a1 /= a0; a2 /= a0;

  // homogeneous basis and sub-block state map (double)
  double al[16], bt[16];
  al[0] = 1.0; bt[0] = 0.0; al[1] = 0.0; bt[1] = 1.0;
  for (int k = 2; k < 16; ++k) {
    al[k] = -a1 * al[k - 1] - a2 * al[k - 2];
    bt[k] = -a1 * bt[k - 1] - a2 * bt[k - 2];
  }
  HM22 M;
  M.a = -a1 * al[15] + (a1 * a1 - a2) * bt[15];
  M.b = -a2 * al[15] + a1 * a2 * bt[15];
  M.c = -a1 * al[14] + (a1 * a1 - a2) * bt[14];
  M.d = -a2 * al[14] + a1 * a2 * bt[14];
  HM22 M2 = hmul(M, M), M4 = hmul(M2, M2), M8 = hmul(M4, M4);
  HM22 Mc = hmul(M8, M8); // M^16 = chunk map
  HM22 C2 = hmul(Mc, Mc), C4 = hmul(C2, C2), C8 = hmul(C4, C4),
       C16 = hmul(C8, C8);

  Params P;
  P.a1 = (float)a1; P.a2 = (float)a2; P.b1 = (float)b1; P.b2 = (float)b2;
  P.al14 = (float)al[14]; P.al15 = (float)al[15];
  P.bt14 = (float)bt[14]; P.bt15 = (float)bt[15];
  put(P.M1, M); put(P.M2, M2); put(P.M4, M4); put(P.M8, M8);
  put(P.C1, Mc); put(P.C2, C2); put(P.C4, C4); put(P.C8, C8); put(P.C16, C16);
  for (int k = 0; k < 16; ++k) {
    P.al[k] = (float)al[k];
    P.bt[k] = (float)bt[k];
  }

  const int threads = 256;                       // 8 waves/block
  const int blocks = (BB * NCH * 32) / threads;  // exactly 120000 waves

  biquad_init<<<1, 32, 0, stream>>>(aopTab, b0, b1, b2, a1, a2);
  biquad_pass13<<<blocks, threads, 0, stream>>>(x, y, dchunk, zin, aopTab, 0, P);
  biquad_pass2<<<BB, 32, 0, stream>>>(dchunk, zin, P);
  biquad_pass13<<<blocks, threads, 0, stream>>>(x, y, dchunk, zin, aopTab, 1, P);
}